// OnlineHadamard_56959856280257
// MI455X (gfx1250) — compile-verified
//
#include <hip/hip_runtime.h>

typedef float v2f __attribute__((ext_vector_type(2)));
typedef float v8f __attribute__((ext_vector_type(8)));

namespace {

constexpr int HDIM = 4096;          // 16 * 16 * 16
constexpr int ROWS_PER_BLOCK = 4;   // one row per wave32
constexpr int THREADS = 32 * ROWS_PER_BLOCK;

// One radix-16 FWHT stage over a 4096-float LDS row.
// Element address: b*BS + N*CS + k*KS   (b = block, N = column-in-block,
// k = contracted index).  Stage1 (k): (256,16,1), Stage2 (j): (256,1,16),
// Stage3 (i): (16,1,256).
template <int BS, int CS, int KS>
__device__ __forceinline__ void fwht_stage(float* __restrict__ Lrow,
                                           const v2f (&A)[4],
                                           int N, int hi) {
  for (int b = 0; b < 16; ++b) {
    const int base = b * BS + N * CS;
    v8f acc = {0.f, 0.f, 0.f, 0.f, 0.f, 0.f, 0.f, 0.f};
#pragma unroll
    for (int c = 0; c < 4; ++c) {
      // B operand 4x16 f32: lane holds (K = c*4 + 2*hi + v, col N) for v=0,1
      v2f Bv;
      Bv.x = Lrow[base + (4 * c + 2 * hi + 0) * KS];
      Bv.y = Lrow[base + (4 * c + 2 * hi + 1) * KS];
      // D(16x16) += H16[:,4c:4c+4] (16x4) x B (4x16)
      acc = __builtin_amdgcn_wmma_f32_16x16x4_f32(false, A[c], false, Bv,
                                                  (short)0, acc, false, false);
    }
    // C/D layout: VGPR v, lanes 0-15 -> M=v, lanes 16-31 -> M=v+8; col N
#pragma unroll
    for (int v = 0; v < 8; ++v) {
      Lrow[base + (v + 8 * hi) * KS] = acc[v];
    }
  }
}

__global__ __launch_bounds__(THREADS) void fwht4096_wmma_kernel(
    const float* __restrict__ x, float* __restrict__ y, int nrows) {
  __shared__ float lds[ROWS_PER_BLOCK * HDIM];  // 64 KB: 5 WGs / 320KB WGP

  const int lane = threadIdx.x & 31;
  const int wave = threadIdx.x >> 5;
  const int row  = blockIdx.x * ROWS_PER_BLOCK + wave;
  if (row >= nrows) return;  // wave-uniform: EXEC stays all-ones for WMMA

  float* Lrow = lds + wave * HDIM;

  // ---------------- global -> LDS, coalesced b128 ----------------------
  const float4* __restrict__ src4 =
      reinterpret_cast<const float4*>(x + (size_t)row * HDIM);
  float4* lds4 = reinterpret_cast<float4*>(Lrow);
#pragma unroll
  for (int t = 0; t < HDIM / 4 / 32; ++t)
    lds4[t * 32 + lane] = src4[t * 32 + lane];

  // ---------------- H16 column slices as WMMA A operands ---------------
  // A 16x4 f32 layout: lane L holds (M = L%16, K = vreg + 2*(L/16)).
  // H16[M][K] = (-1)^popcount(M & K)  (Sylvester ordering).
  const int N  = lane & 15;
  const int hi = lane >> 4;
  v2f A[4];
#pragma unroll
  for (int c = 0; c < 4; ++c) {
#pragma unroll
    for (int v = 0; v < 2; ++v) {
      const int K = 4 * c + 2 * hi + v;
      A[c][v] = (__builtin_popcount(N & K) & 1) ? -1.0f : 1.0f;
    }
  }

  // ---------------- three radix-16 stages (wave-private, no barriers) --
  fwht_stage<256, 16, 1>(Lrow, A, N, hi);   // contract k (stride 1)
  fwht_stage<256, 1, 16>(Lrow, A, N, hi);   // contract j (stride 16)
  fwht_stage<16, 1, 256>(Lrow, A, N, hi);   // contract i (stride 256)

  // ---------------- LDS -> global, coalesced b128 ----------------------
  float4* __restrict__ dst4 =
      reinterpret_cast<float4*>(y + (size_t)row * HDIM);
#pragma unroll
  for (int t = 0; t < HDIM / 4 / 32; ++t)
    dst4[t * 32 + lane] = lds4[t * 32 + lane];
}

}  // namespace

extern "C" void kernel_launch(void* const* d_in, const int* in_sizes, int n_in,
                              void* d_out, int out_size, void* d_ws, size_t ws_size,
                              hipStream_t stream) {
  const float* x = (const float*)d_in[0];
  float* y = (float*)d_out;
  const int nrows = in_sizes[0] / HDIM;                     // 8192
  const int blocks = (nrows + ROWS_PER_BLOCK - 1) / ROWS_PER_BLOCK;
  fwht4096_wmma_kernel<<<blocks, THREADS, 0, stream>>>(x, y, nrows);
}